// HSFil_62508954026541
// MI455X (gfx1250) — compile-verified
//
#include <hip/hip_runtime.h>

typedef __attribute__((ext_vector_type(2))) float v2f;
typedef __attribute__((ext_vector_type(8))) float v8f;

__device__ __forceinline__ float log_sigmoid_stable(float s) {
    // logsigmoid(s) = min(s,0) - log1p(exp(-|s|))   (numerically stable)
    return fminf(s, 0.0f) - log1pf(expf(-fabsf(s)));
}

// One wave (32 lanes) per sample. WMMA f32 16x16x4:
//   A = 16 gathered decision rows (M = path position), K-chunked over D
//   B = x[b] chunk broadcast into all 16 N columns
//   D-tile: every column holds the 16 scores.
__global__ void __launch_bounds__(256)
hs_loss_kernel(const float* __restrict__ x,
               const int*   __restrict__ t,
               const float* __restrict__ decs,
               const int*   __restrict__ paths,
               const int*   __restrict__ lens,
               float*       __restrict__ partials,
               int Bn, int D, int MAXP)
{
    __shared__ float wpart[8];
    const int lane = threadIdx.x & 31;
    const int wave = threadIdx.x >> 5;
    const int b    = blockIdx.x * 8 + wave;

    float wave_total = 0.0f;
    if (b < Bn) {                               // wave-uniform guard
        const int tt    = t[b];
        const int len   = lens[tt];
        const int m     = lane & 15;            // path position within tile
        const int khalf = lane >> 4;            // which K-pair this lane feeds
        const int cofs  = 2 * khalf;
        const float* __restrict__ xrow = x + (size_t)b * D;
        const int ntiles = (len > 16) ? 2 : 1;  // wave-uniform

        // Prefetch tile-0 gathered rows into cache hierarchy.
        {
            const int r0 = paths[(size_t)tt * MAXP + m];
            __builtin_prefetch(decs + (size_t)r0 * D, 0, 1);
        }

        float persum = 0.0f;
        for (int tile = 0; tile < ntiles; ++tile) {
            const int r = paths[(size_t)tt * MAXP + tile * 16 + m];
            const float* __restrict__ wrow = decs + (size_t)r * D;
            if (tile + 1 < ntiles) {
                const int rn = paths[(size_t)tt * MAXP + 16 + m];
                __builtin_prefetch(decs + (size_t)rn * D, 0, 1);
            }

            v8f acc = {0.f, 0.f, 0.f, 0.f, 0.f, 0.f, 0.f, 0.f};
            #pragma unroll 4
            for (int k0 = 0; k0 < D; k0 += 4) {
                // A lane data: decs[r][k0+2*khalf .. +1]  (K pair per half-wave)
                // B lane data: x[b][k0+2*khalf .. +1]     (same column, broadcast N)
                v2f a  = *(const v2f*)(wrow + k0 + cofs);
                v2f bx = *(const v2f*)(xrow + k0 + cofs);
                acc = __builtin_amdgcn_wmma_f32_16x16x4_f32(
                        /*neg_a=*/false, a, /*neg_b=*/false, bx,
                        /*c_mod=*/(short)0, acc,
                        /*reuse_a=*/false, /*reuse_b=*/false);
            }

            // D-tile layout: acc[j] holds M=j (lanes 0-15) / M=j+8 (lanes 16-31),
            // identical across the 16 columns. Mask invalid path entries.
            #pragma unroll
            for (int j = 0; j < 8; ++j) {
                const int mg = tile * 16 + 8 * khalf + j;
                const float ls = log_sigmoid_stable(acc[j]);
                persum += (mg < len) ? ls : 0.0f;
            }
        }
        // lanes 0-15 all hold the M=0..7 sum, lanes 16-31 the M=8..15 sum.
        wave_total = __shfl(persum, 0, 32) + __shfl(persum, 16, 32);
    }

    if (lane == 0) wpart[wave] = wave_total;
    __syncthreads();
    if (threadIdx.x == 0) {
        float s = 0.0f;
        #pragma unroll
        for (int i = 0; i < 8; ++i) s += wpart[i];
        partials[blockIdx.x] = s;
    }
}

// Deterministic fixed-order final reduction: loss = -sum(partials) / B.
__global__ void __launch_bounds__(256)
hs_reduce_kernel(const float* __restrict__ partials, int nparts, float invB,
                 float* __restrict__ out)
{
    __shared__ float sm[256];
    float s = 0.0f;
    for (int i = threadIdx.x; i < nparts; i += 256) s += partials[i];
    sm[threadIdx.x] = s;
    __syncthreads();
    #pragma unroll
    for (int off = 128; off > 0; off >>= 1) {
        if (threadIdx.x < off) sm[threadIdx.x] += sm[threadIdx.x + off];
        __syncthreads();
    }
    if (threadIdx.x == 0) out[0] = -sm[0] * invB;
}

extern "C" void kernel_launch(void* const* d_in, const int* in_sizes, int n_in,
                              void* d_out, int out_size, void* d_ws, size_t ws_size,
                              hipStream_t stream)
{
    const float* x     = (const float*)d_in[0];
    const int*   t     = (const int*)  d_in[1];
    const float* decs  = (const float*)d_in[2];
    const int*   paths = (const int*)  d_in[3];
    const int*   lens  = (const int*)  d_in[4];

    const int Bn   = in_sizes[1];               // 8192
    const int D    = in_sizes[0] / Bn;          // 1024
    const int VOC  = in_sizes[4];               // 50257
    const int MAXP = in_sizes[3] / VOC;         // 32

    float* partials = (float*)d_ws;
    const int nblocks = (Bn + 7) / 8;           // 8 waves/block, 1 sample/wave

    hs_loss_kernel<<<nblocks, 256, 0, stream>>>(x, t, decs, paths, lens,
                                                partials, Bn, D, MAXP);
    hs_reduce_kernel<<<1, 256, 0, stream>>>(partials, nblocks, 1.0f / (float)Bn,
                                            (float*)d_out);
}